// TensorDecomposition_13726715478497
// MI455X (gfx1250) — compile-verified
//
#include <hip/hip_runtime.h>
#include <cstdint>

#define DIM    300
#define CCH    16
#define TEXELS (DIM * DIM)          // 90000 texels per plane
#define LINE_ELEMS (3 * DIM * CCH)  // 14400 floats = 57.6 KB in LDS

#if defined(__has_builtin)
# if __has_builtin(__builtin_amdgcn_global_load_async_to_lds_b32)
#  define HAVE_ASYNC_LDS 1
# endif
# if __has_builtin(__builtin_amdgcn_s_wait_asynccnt)
#  define HAVE_WAIT_ASYNC 1
# endif
#endif

typedef __attribute__((address_space(1))) int* gptr_i32;
typedef __attribute__((address_space(3))) int* lptr_i32;

__device__ __forceinline__ float4 ld4(const float* p) {
    return *reinterpret_cast<const float4*>(p);
}
__device__ __forceinline__ void st4(float* p, const float4& v) {
    *reinterpret_cast<float4*>(p) = v;
}
__device__ __forceinline__ float4 lerp4(const float4& a, const float4& b, float w) {
    float4 r;
    r.x = fmaf(w, b.x - a.x, a.x);
    r.y = fmaf(w, b.y - a.y, a.y);
    r.z = fmaf(w, b.z - a.z, a.z);
    r.w = fmaf(w, b.w - a.w, a.w);
    return r;
}
__device__ __forceinline__ float4 mul4(const float4& a, const float4& b) {
    float4 r; r.x = a.x*b.x; r.y = a.y*b.y; r.z = a.z*b.z; r.w = a.w*b.w; return r;
}

// Map normalized coord t in [-1,1] to taps (i0,i1) and weight w, matching the
// reference: i0 = clip(floor(x),0,dim-1), i1 = clip(i0+1,0,dim-1), w = x - floor(x).
__device__ __forceinline__ void texw(float t, int& i0, int& i1, float& w) {
    float x  = (t + 1.0f) * 0.5f * (float)(DIM - 1);
    float xf = floorf(x);
    w = x - xf;
    int i = (int)xf;
    i0 = min(max(i, 0), DIM - 1);
    i1 = min(i0 + 1, DIM - 1);
}

// ---------------------------------------------------------------------------
// Stage the three lines into LDS, transposed to channels-last:
// sLine[(line*300 + pos)*16 + c]. With the CDNA5 async-to-LDS path each lane
// gets an independent LDS destination, so the transpose is free in flight.
// ---------------------------------------------------------------------------
__device__ __forceinline__ void stage_lines(
    float* sLine, const float* lx, const float* ly, const float* lz)
{
    const float* lsrc[3] = { lx, ly, lz };
    for (int g = threadIdx.x; g < LINE_ELEMS; g += blockDim.x) {
        int l   = g / (DIM * CCH);
        int r   = g - l * (DIM * CCH);
        int c   = r / DIM;
        int pos = r - c * DIM;
        const float* gp = lsrc[l] + c * DIM + pos;           // coalesced global read
        float*       lp = &sLine[(l * DIM + pos) * CCH + c]; // scattered LDS write
#if defined(HAVE_ASYNC_LDS)
        __builtin_amdgcn_global_load_async_to_lds_b32(
            (gptr_i32)(const_cast<float*>(gp)),
            (lptr_i32)(lp),
            0, 0);
#else
        *lp = *gp;
#endif
    }
#if defined(HAVE_ASYNC_LDS)
# if defined(HAVE_WAIT_ASYNC)
    __builtin_amdgcn_s_wait_asynccnt(0);
# else
    asm volatile("s_wait_asynccnt 0" ::: "memory");
# endif
#endif
    __syncthreads();
}

// Combo table: output o uses plane PK[o] (== coords_plane row PK[o]) and
// line o (== coords_line row o).
//   o=0 (out_x): plane_yz (1) * line_x (0)
//   o=1 (out_y): plane_xz (2) * line_y (1)
//   o=2 (out_z): plane_xy (0) * line_z (2)
__constant__ int PK_TBL[3] = { 1, 2, 0 };

// ---------------------------------------------------------------------------
// Fast path: channels-last planes in d_ws, lines in LDS.
// One sample per thread per grid-stride step; all loads/stores are B64/B128.
// (Defined first so the disasm snippet shows the async-to-LDS staging.)
// ---------------------------------------------------------------------------
__global__ __launch_bounds__(256) void k_fused_ws(
    const float* __restrict__ cP, const float* __restrict__ cL,
    const float* __restrict__ ws,
    const float* __restrict__ lx, const float* __restrict__ ly,
    const float* __restrict__ lz,
    float* __restrict__ out, int N)
{
    __shared__ __align__(16) float sLine[LINE_ELEMS];
    stage_lines(sLine, lx, ly, lz);

    int tid    = blockIdx.x * blockDim.x + threadIdx.x;
    int stride = gridDim.x * blockDim.x;
    for (int n = tid; n < N; n += stride) {
        int npf = n + stride;
        if (npf < N) {
            __builtin_prefetch(cP + (size_t)npf * 2, 0, 3);
            __builtin_prefetch(cL + (size_t)npf * 2, 0, 3);
        }
#pragma unroll
        for (int o = 0; o < 3; ++o) {
            const int pk = PK_TBL[o];
            const float2 cp = *reinterpret_cast<const float2*>(cP + ((size_t)pk * N + n) * 2);
            int ix0, ix1, iy0, iy1; float wx, wy;
            texw(cp.x, ix0, ix1, wx);
            texw(cp.y, iy0, iy1, wy);
            const float* pb  = ws + (size_t)pk * (TEXELS * CCH);
            const float* p00 = pb + (iy0 * DIM + ix0) * CCH;
            const float* p01 = pb + (iy0 * DIM + ix1) * CCH;
            const float* p10 = pb + (iy1 * DIM + ix0) * CCH;
            const float* p11 = pb + (iy1 * DIM + ix1) * CCH;

            const float2 cl = *reinterpret_cast<const float2*>(cL + ((size_t)o * N + n) * 2);
            int l0i, l1i; float wl;
            texw(cl.y, l0i, l1i, wl);   // lines interpolate along the y coord (W==1)
            const float* l0 = &sLine[(o * DIM + l0i) * CCH];
            const float* l1 = &sLine[(o * DIM + l1i) * CCH];

            float* po = out + ((size_t)o * N + n) * CCH;
#pragma unroll
            for (int cg = 0; cg < 4; ++cg) {
                float4 v00 = ld4(p00 + cg * 4);
                float4 v01 = ld4(p01 + cg * 4);
                float4 v10 = ld4(p10 + cg * 4);
                float4 v11 = ld4(p11 + cg * 4);
                float4 f = lerp4(lerp4(v00, v01, wx), lerp4(v10, v11, wx), wy);
                float4 L = lerp4(ld4(l0 + cg * 4), ld4(l1 + cg * 4), wl);
                st4(po + cg * 4, mul4(f, L));
            }
        }
    }
}

// ---------------------------------------------------------------------------
// Repack the three (16,300,300) planes into channels-last [plane][texel][16]
// so each bilinear tap is one 64-byte (4x float4) contiguous read.
// Reads are lane-coalesced (consecutive texels per lane); writes are B128.
// ---------------------------------------------------------------------------
__global__ __launch_bounds__(256) void k_transpose_planes(
    const float* __restrict__ pxy, const float* __restrict__ pyz,
    const float* __restrict__ pxz, float* __restrict__ ws)
{
    int t = blockIdx.x * blockDim.x + threadIdx.x;
    if (t >= 3 * TEXELS) return;
    int p  = t / TEXELS;
    int t2 = t - p * TEXELS;
    const float* src = (p == 0) ? pxy : (p == 1) ? pyz : pxz;
    float v[CCH];
#pragma unroll
    for (int c = 0; c < CCH; ++c) v[c] = src[c * TEXELS + t2];
    float* dst = ws + (size_t)t * CCH;
#pragma unroll
    for (int cg = 0; cg < 4; ++cg)
        st4(dst + cg * 4, make_float4(v[cg*4+0], v[cg*4+1], v[cg*4+2], v[cg*4+3]));
}

// ---------------------------------------------------------------------------
// Fallback path if the workspace is too small for the channels-last planes:
// gather straight from the (16,300,300) layout with scalar loads.
// ---------------------------------------------------------------------------
__global__ __launch_bounds__(256) void k_fused_direct(
    const float* __restrict__ cP, const float* __restrict__ cL,
    const float* __restrict__ pxy, const float* __restrict__ pyz,
    const float* __restrict__ pxz,
    const float* __restrict__ lx, const float* __restrict__ ly,
    const float* __restrict__ lz,
    float* __restrict__ out, int N)
{
    __shared__ __align__(16) float sLine[LINE_ELEMS];
    stage_lines(sLine, lx, ly, lz);

    const float* psrc[3] = { pxy, pyz, pxz };
    int tid    = blockIdx.x * blockDim.x + threadIdx.x;
    int stride = gridDim.x * blockDim.x;
    for (int n = tid; n < N; n += stride) {
#pragma unroll
        for (int o = 0; o < 3; ++o) {
            const int pk = PK_TBL[o];
            const float2 cp = *reinterpret_cast<const float2*>(cP + ((size_t)pk * N + n) * 2);
            int ix0, ix1, iy0, iy1; float wx, wy;
            texw(cp.x, ix0, ix1, wx);
            texw(cp.y, iy0, iy1, wy);
            const float* pb = psrc[pk];
            int r00 = iy0 * DIM + ix0, r01 = iy0 * DIM + ix1;
            int r10 = iy1 * DIM + ix0, r11 = iy1 * DIM + ix1;

            const float2 cl = *reinterpret_cast<const float2*>(cL + ((size_t)o * N + n) * 2);
            int l0i, l1i; float wl;
            texw(cl.y, l0i, l1i, wl);
            const float* l0 = &sLine[(o * DIM + l0i) * CCH];
            const float* l1 = &sLine[(o * DIM + l1i) * CCH];

            float* po = out + ((size_t)o * N + n) * CCH;
#pragma unroll
            for (int c = 0; c < CCH; ++c) {
                const float* pc = pb + c * TEXELS;
                float h0 = fmaf(wx, pc[r01] - pc[r00], pc[r00]);
                float h1 = fmaf(wx, pc[r11] - pc[r10], pc[r10]);
                float f  = fmaf(wy, h1 - h0, h0);
                float L  = fmaf(wl, l1[c] - l0[c], l0[c]);
                po[c] = f * L;
            }
        }
    }
}

extern "C" void kernel_launch(void* const* d_in, const int* in_sizes, int n_in,
                              void* d_out, int out_size, void* d_ws, size_t ws_size,
                              hipStream_t stream) {
    const float* cP  = (const float*)d_in[0];   // coords_plane (3,1,N,1,2)
    const float* cL  = (const float*)d_in[1];   // coords_line  (3,1,N,1,2)
    const float* pxy = (const float*)d_in[2];   // (1,16,300,300)
    const float* pyz = (const float*)d_in[3];
    const float* pxz = (const float*)d_in[4];
    const float* lx  = (const float*)d_in[5];   // (1,16,300,1)
    const float* ly  = (const float*)d_in[6];
    const float* lz  = (const float*)d_in[7];
    float* out = (float*)d_out;

    const int N = in_sizes[0] / 6;              // (3*N*2) elements
    const size_t ws_need = (size_t)3 * TEXELS * CCH * sizeof(float); // 17.28 MB

    const int threads = 256;
    const int fused_blocks = 1024;              // grid-stride, ~4 samples/thread

    if (ws_size >= ws_need) {
        float* ws = (float*)d_ws;
        int tblocks = (3 * TEXELS + threads - 1) / threads;
        k_transpose_planes<<<tblocks, threads, 0, stream>>>(pxy, pyz, pxz, ws);
        k_fused_ws<<<fused_blocks, threads, 0, stream>>>(cP, cL, ws, lx, ly, lz, out, N);
    } else {
        k_fused_direct<<<fused_blocks, threads, 0, stream>>>(cP, cL, pxy, pyz, pxz,
                                                             lx, ly, lz, out, N);
    }
}